// CoarseDirectionReducer_9483287789866
// MI455X (gfx1250) — compile-verified
//
#include <hip/hip_runtime.h>

typedef __attribute__((ext_vector_type(2))) float v2f;
typedef __attribute__((ext_vector_type(8))) float v8f;

// Inverse of GROUP_IDX: fine-channel k -> (group, slot). Each k appears exactly once.
// GROUP_IDX = {{17,23,18},{16,21,22},{15,19,20},{12,13,9},{11,10,14},{8,4,3},{7,2,1},{6,0,5}}
__device__ __constant__ int cK2G[24] = {7,6,6,5,5,7,7,6,5,3,4,4,3,3,4,2,1,0,0,2,2,1,1,0};
__device__ __constant__ int cK2J[24] = {1,2,1,2,1,2,0,0,0,2,1,0,0,1,2,0,0,0,2,1,2,1,2,1};

// Fixed problem shape (B=4, NC=4, H=W=256) -> all strides become compile-time
// constants that fit the VGLOBAL 24-bit signed immediate offset (+-8MB):
//   max load  imm = 21*HW*4 + 7*64 = 5,505,472 B
//   max store imm =  7*HW*4 + 7*64 = 1,835,456 B
constexpr int HW     = 256 * 256;   // 65536
constexpr int PLANES = 4 * 4;       // B*NC
constexpr int TPP    = HW / 16;     // 4096 tiles per plane (power of two)
#define TPW 8                       // 16-pixel tiles per wave

__global__ __launch_bounds__(256) void coarse_dir_reduce_wmma(
    const float* __restrict__ x,       // (PLANES, 24, HW)
    const float* __restrict__ logits,  // (8, 3)
    float* __restrict__ out) {         // (PLANES, 8, HW)

  const int lane = threadIdx.x & 31;
  const int row  = lane & 15;          // N (pixel-in-tile) / M (group) index
  const int hi   = lane >> 4;          // 0: K pair {0,1}, 1: K pair {2,3}

  const int wid = blockIdx.x * (blockDim.x >> 5) + (threadIdx.x >> 5);
  const int t0  = wid * TPW;
  if (t0 >= PLANES * TPP) return;

  // ---- softmax(logits) per group, all registers (fully unrolled) ----
  float sm[8][3];
#pragma unroll
  for (int g = 0; g < 8; ++g) {
    float l0 = logits[g * 3 + 0];
    float l1 = logits[g * 3 + 1];
    float l2 = logits[g * 3 + 2];
    float m  = fmaxf(l0, fmaxf(l1, l2));
    float e0 = __expf(l0 - m), e1 = __expf(l1 - m), e2 = __expf(l2 - m);
    float inv = 1.0f / (e0 + e1 + e2);
    sm[g][0] = e0 * inv; sm[g][1] = e1 * inv; sm[g][2] = e2 * inv;
  }

  // Wmat[row][k] = (row == group(k)) ? softmax weight of k : 0  (rows 8..15 -> 0)
  auto wval = [&](int k) -> float {
    int g = cK2G[k], j = cK2J[k];
    return (row == g) ? sm[g][j] : 0.0f;
  };

  // ---- A fragments: six 16x4 f32 chunks of the padded weight matrix ----
  // ISA layout (A 16x4 f32): VGPR v, lane L -> M = L%16, K = v + 2*(L>=16)
  v2f a[6];
#pragma unroll
  for (int c = 0; c < 6; ++c) {
    float x0 = wval(4 * c + 0), x1 = wval(4 * c + 1);
    float x2 = wval(4 * c + 2), x3 = wval(4 * c + 3);
    a[c].x = hi ? x2 : x0;
    a[c].y = hi ? x3 : x1;
  }

  // Plane uniform per wave (TPP multiple of TPW). One per-lane base address for
  // input (folds in the lane-dependent 2*hi*HW channel term) and one for
  // output; every access below is base + compile-time immediate offset.
  const int plane = t0 >> 12;              // t0 / TPP
  const int pix0  = (t0 & (TPP - 1)) * 16; // tile0 * 16
  const float* bp = x   + ((size_t)plane * 24 + 2 * hi) * HW + pix0 + row;
  float*       op = out + ((size_t)plane * 8) * HW + pix0 + row;

#pragma unroll
  for (int t = 0; t < TPW; ++t) {
    v8f acc = {};
#pragma unroll
    for (int c = 0; c < 6; ++c) {
      // B fragment (4x16 f32): VGPR v, lane L -> K = v + 2*(L>=16), N = L%16
      v2f b;
      b.x = bp[(4 * c + 0) * HW + t * 16];
      b.y = bp[(4 * c + 1) * HW + t * 16];
      acc = __builtin_amdgcn_wmma_f32_16x16x4_f32(
          /*neg_a=*/false, a[c], /*neg_b=*/false, b,
          /*c_mod=*/(short)0, acc, /*reuse_a=*/false, /*reuse_b=*/false);
    }
    // D layout: VGPR g, lanes 0-15 hold group g, col = lane. Lanes 16-31 hold
    // the zero-padded rows 8..15 -> masked half-wave NT store.
    if (lane < 16) {
#pragma unroll
      for (int g = 0; g < 8; ++g) {
        __builtin_nontemporal_store(acc[g], &op[g * HW + t * 16]);
      }
    }
  }
}

extern "C" void kernel_launch(void* const* d_in, const int* in_sizes, int n_in,
                              void* d_out, int out_size, void* d_ws, size_t ws_size,
                              hipStream_t stream) {
  const float* x      = (const float*)d_in[0];
  const float* logits = (const float*)d_in[1];
  float* out          = (float*)d_out;

  const int totalTiles    = PLANES * TPP;                 // 65536
  const int wavesPerBlock = 8;                            // 256 threads, wave32
  const int waves         = (totalTiles + TPW - 1) / TPW; // 8192
  const int blocks        = (waves + wavesPerBlock - 1) / wavesPerBlock; // 1024

  coarse_dir_reduce_wmma<<<blocks, 256, 0, stream>>>(x, logits, out);
}